// KBestDetector_39616778338615
// MI455X (gfx1250) — compile-verified
//
#include <hip/hip_runtime.h>
#include <math.h>

// CDNA5 / gfx1250: wave32, WMMA 16x16x4 f32 for the Gram matrix.
typedef __attribute__((ext_vector_type(2))) float v2f;
typedef __attribute__((ext_vector_type(8))) float v8f;

#define N      16      // 2 * NUM_STREAMS
#define KBEST  64
#define MPAM   4
#define NCMAX  (KBEST * MPAM)

// PAM points: (2*[0..3]-3) / (std_ddof1 * sqrt(2)) = [-3,-1,1,3]/sqrt(40/3)
__constant__ float c_pam[MPAM] = {-0.8215838362577491f, -0.27386127875258304f,
                                   0.27386127875258304f,  0.8215838362577491f};

// Parallel in-place lower Cholesky of a 16x16 SPD matrix in LDS (wave32).
__device__ __forceinline__ void chol16(float* M, int lane) {
  for (int k = 0; k < N; ++k) {
    if (lane == 0) M[k*N + k] = sqrtf(M[k*N + k]);
    __syncthreads();
    float dk = M[k*N + k];
    if (lane > k && lane < N) M[lane*N + k] /= dk;
    __syncthreads();
    for (int idx = lane; idx < N*N; idx += 32) {
      int i = idx >> 4, j = idx & 15;
      if (i > k && j > k && j <= i) M[idx] -= M[i*N + k] * M[j*N + k];
    }
    __syncthreads();
  }
}

__global__ __launch_bounds__(32)
void kbest_detector(const float* __restrict__ g_yre, const float* __restrict__ g_yim,
                    const float* __restrict__ g_hre, const float* __restrict__ g_him,
                    const float* __restrict__ g_sre, const float* __restrict__ g_sim,
                    float* __restrict__ g_out)
{
  __shared__ float Smat[N*N];          // real S, then its Cholesky L (lower)
  __shared__ float Hmat[N*N];          // real H
  __shared__ float Amat[N*N];          // A = L^{-1} H
  __shared__ float Gmat[N*N];          // G = A^T A  (from WMMA)
  __shared__ float Gp[N*N];            // permuted G, then its Cholesky Lg (lower)
  __shared__ float Rmat[N*N];          // R = Lg^T (upper)
  __shared__ float yvec[N], wvec[N], bvec[N], bpv[N], zvec[N], nrm[N];
  __shared__ int   perm[N];
  __shared__ float dists[KBEST];
  __shared__ float symbuf[2][KBEST*N]; // survivor symbols, indexed by row si
  __shared__ float candD[NCMAX];

  const int b    = blockIdx.x;
  const int lane = threadIdx.x;        // 0..31, one wave per batch element

  // ---------- build real-valued S (16x16), H (16x16), y (16) ----------
  {
    const float* sre = g_sre + (size_t)b * 64;
    const float* sim = g_sim + (size_t)b * 64;
    const float* hre = g_hre + (size_t)b * 64;
    const float* him = g_him + (size_t)b * 64;
    for (int idx = lane; idx < 64; idx += 32) {
      int i = idx >> 3, j = idx & 7;
      float sr = sre[idx], si = sim[idx];
      Smat[i*N + j]         =  sr;   // (global 0.5 scale in the reference only
      Smat[i*N + j + 8]     = -si;   //  scales all distances uniformly -> same argmin)
      Smat[(i+8)*N + j]     =  si;
      Smat[(i+8)*N + j + 8] =  sr;
      float hr = hre[idx], hi = him[idx];
      Hmat[i*N + j]         =  hr;
      Hmat[i*N + j + 8]     = -hi;
      Hmat[(i+8)*N + j]     =  hi;
      Hmat[(i+8)*N + j + 8] =  hr;
    }
    if (lane < 8) {
      yvec[lane]     = g_yre[(size_t)b*8 + lane];
      yvec[lane + 8] = g_yim[(size_t)b*8 + lane];
    }
  }
  __syncthreads();

  // ---------- S = L L^T ----------
  chol16(Smat, lane);

  // ---------- A = L^{-1} H (lanes 0..15: one column each), w = L^{-1} y (lane 16) ----------
  if (lane <= N) {
    float x[N];
    #pragma unroll
    for (int i = 0; i < N; ++i) {
      float v = (lane < N) ? Hmat[i*N + lane] : yvec[i];
      #pragma unroll
      for (int j = 0; j < i; ++j) v -= Smat[i*N + j] * x[j];
      x[i] = v / Smat[i*N + i];
    }
    #pragma unroll
    for (int i = 0; i < N; ++i) {
      if (lane < N) Amat[i*N + lane] = x[i];
      else          wvec[i] = x[i];
    }
  }
  __syncthreads();

  // ---------- G = A^T A via 4x V_WMMA_F32_16X16X4_F32 ----------
  // For chunk kk, both the A-operand (rows of A^T) and B-operand (rows of A)
  // have the same lane/VGPR image: lane L<16 holds {A[4kk+0][L], A[4kk+1][L]},
  // lane L+16 holds {A[4kk+2][L], A[4kk+3][L]}.  EXEC is all-ones here.
  {
    v8f acc = {0.f, 0.f, 0.f, 0.f, 0.f, 0.f, 0.f, 0.f};
    const int c  = lane & 15;
    const int hk = (lane >> 4) << 1;  // 0 or 2
    #pragma unroll
    for (int kk = 0; kk < 4; ++kk) {
      int r0 = 4*kk + hk;
      v2f ab;
      ab.x = Amat[r0*N + c];
      ab.y = Amat[(r0 + 1)*N + c];
      acc = __builtin_amdgcn_wmma_f32_16x16x4_f32(false, ab, false, ab,
                                                  (short)0, acc, false, false);
    }
    // C/D layout: vgpr r, lanes 0-15 -> G[r][lane]; lanes 16-31 -> G[r+8][lane-16]
    const int rowbase = (lane >> 4) * 8;
    #pragma unroll
    for (int r = 0; r < 8; ++r)
      Gmat[(rowbase + r)*N + c] = acc[r];
  }
  // b = A^T w
  if (lane < N) {
    float a = 0.f;
    #pragma unroll
    for (int i = 0; i < N; ++i) a += Amat[i*N + lane] * wvec[i];
    bvec[lane] = a;
  }
  __syncthreads();

  // ---------- column ordering by descending whitened norm (stable) ----------
  if (lane < N) nrm[lane] = Gmat[lane*N + lane];
  __syncthreads();
  if (lane < N) {
    float nc = nrm[lane];
    int rank = 0;
    #pragma unroll
    for (int j = 0; j < N; ++j) {
      float nj = nrm[j];
      rank += (nj > nc) || (nj == nc && j < lane);
    }
    perm[rank] = lane;
  }
  __syncthreads();

  // ---------- permute Gram + rhs ----------
  for (int idx = lane; idx < N*N; idx += 32) {
    int i = idx >> 4, j = idx & 15;
    Gp[idx] = Gmat[perm[i]*N + perm[j]];
  }
  if (lane < N) bpv[lane] = bvec[perm[lane]];
  __syncthreads();

  // ---------- R^T R = Gp :  Gp -> Lg (lower),  R = Lg^T ----------
  chol16(Gp, lane);

  // z = Lg^{-1} b_perm  (equivalent to Q^T y up to row signs -> same detection)
  if (lane == 0) {
    float zz[N];
    #pragma unroll
    for (int i = 0; i < N; ++i) {
      float v = bpv[i];
      #pragma unroll
      for (int j = 0; j < i; ++j) v -= Gp[i*N + j] * zz[j];
      zz[i] = v / Gp[i*N + i];
      zvec[i] = zz[i];
    }
  }
  for (int idx = lane; idx < N*N; idx += 32) {
    int i = idx >> 4, j = idx & 15;
    Rmat[idx] = (j >= i) ? Gp[j*N + i] : 0.f;
  }
  if (lane == 0) dists[0] = 0.f;
  __syncthreads();

  // ---------- K-best tree search: 16 levels, expand x4, keep 64 smallest ----------
  int P = 1, cur = 0;
  for (int l = 0; l < N; ++l) {
    const int si    = N - 1 - l;
    const int ncand = P * MPAM;
    __syncthreads();  // survivors of previous level visible
    const float rss = Rmat[si*N + si];
    const float zs  = zvec[si];
    for (int cidx = lane; cidx < ncand; cidx += 32) {
      const int p = cidx >> 2, q = cidx & 3;
      const float* sp = &symbuf[cur][p*N];
      float partial = 0.f;
      for (int j = si + 1; j < N; ++j) partial += Rmat[si*N + j] * sp[j];
      float t = zs - partial - rss * c_pam[q];
      candD[cidx] = dists[p] + t * t;
    }
    __syncthreads();
    const int knext = (ncand < KBEST) ? ncand : KBEST;
    // branchless rank-by-count selection (stable ties like top_k on negated dists)
    for (int cidx = lane; cidx < ncand; cidx += 32) {
      const float dc = candD[cidx];
      int rank = 0;
      for (int cc = 0; cc < ncand; ++cc) {
        const float d2 = candD[cc];
        rank += (d2 < dc) || (d2 == dc && cc < cidx);
      }
      if (rank < knext) {
        const int p = cidx >> 2, q = cidx & 3;
        float*       dst = &symbuf[cur ^ 1][rank*N];
        const float* src = &symbuf[cur][p*N];
        for (int j = si + 1; j < N; ++j) dst[j] = src[j];
        dst[si]     = c_pam[q];
        dists[rank] = dc;
      }
    }
    cur ^= 1;
    P = knext;
  }
  __syncthreads();

  // best survivor is rank 0; symbuf[cur][si] = symbol of permuted row si.
  // reference: out[order[t]] = best-symbol-of-row-t
  if (lane < N) g_out[(size_t)b*N + perm[lane]] = symbuf[cur][lane];
}

extern "C" void kernel_launch(void* const* d_in, const int* in_sizes, int n_in,
                              void* d_out, int out_size, void* d_ws, size_t ws_size,
                              hipStream_t stream) {
  const float* y_re = (const float*)d_in[0];
  const float* y_im = (const float*)d_in[1];
  const float* h_re = (const float*)d_in[2];
  const float* h_im = (const float*)d_in[3];
  const float* s_re = (const float*)d_in[4];
  const float* s_im = (const float*)d_in[5];
  float* out = (float*)d_out;

  const int B = in_sizes[0] / 8;   // y_re is (B, 8)
  kbest_detector<<<B, 32, 0, stream>>>(y_re, y_im, h_re, h_im, s_re, s_im, out);
}